// CustomNaiveGate_Balance_SMoE_11063835755040
// MI455X (gfx1250) — compile-verified
//
#include <hip/hip_runtime.h>
#include <hip/hip_bf16.h>

typedef __attribute__((ext_vector_type(2))) float v2f;
typedef __attribute__((ext_vector_type(4))) float v4f;
typedef __attribute__((ext_vector_type(8))) float v8f;

#define N_TOKENS_C   32768
#define D_MODEL_C    1024
#define N_EXPERT_C   16
#define WAVES_PER_BLOCK 8
#define TOKENS_PER_BLOCK (WAVES_PER_BLOCK * 16)

// ---------------------------------------------------------------------------
// Kernel 0: zero the 32-float workspace (16 expert counts + 16 prob sums).
// ---------------------------------------------------------------------------
__global__ void moe_init_ws(float* __restrict__ ws) {
    if (threadIdx.x < 32) ws[threadIdx.x] = 0.0f;
}

// ---------------------------------------------------------------------------
// Kernel 1: fused gate GEMM (WMMA f32 16x16x4) + top-2 + softmax + loss bins.
// One wave computes one 16-token x 16-expert logits tile.
// ---------------------------------------------------------------------------
__launch_bounds__(256)
__global__ void moe_gate_kernel(const float* __restrict__ inp,   // [N, D]
                                const float* __restrict__ W,     // [E, D]
                                float* __restrict__ out_idx,     // [N, 2] (indices as float)
                                float* __restrict__ out_score,   // [N, 2]
                                float* __restrict__ ws_cnt,      // [16]
                                float* __restrict__ ws_psum)     // [16]
{
    __shared__ float logits_lds[WAVES_PER_BLOCK][16][16];
    __shared__ float cnt_lds[N_EXPERT_C];
    __shared__ float psum_lds[N_EXPERT_C];

    const int tid  = threadIdx.x;
    const int lane = tid & 31;
    const int wave = tid >> 5;

    if (tid < N_EXPERT_C) { cnt_lds[tid] = 0.0f; psum_lds[tid] = 0.0f; }
    __syncthreads();

    const int tile_base = blockIdx.x * TOKENS_PER_BLOCK + wave * 16;
    const int m  = lane & 15;          // A: token row / B: expert col
    const int kh = (lane >> 4) << 1;   // K sub-offset: 0 (lanes 0-15) or 2 (lanes 16-31)

    const float* __restrict__ aptr = inp + (size_t)(tile_base + m) * D_MODEL_C + kh;
    const float* __restrict__ bptr = W   + (size_t)m              * D_MODEL_C + kh;

    // ---- GEMM: 256 x V_WMMA_F32_16X16X4_F32, K walked contiguously ----
    v8f c = {0.f, 0.f, 0.f, 0.f, 0.f, 0.f, 0.f, 0.f};
#pragma unroll 8
    for (int k = 0; k < D_MODEL_C; k += 4) {
        v2f a = *(const v2f*)(aptr + k);   // global_load_b64 (A tile, striped)
        v2f b = *(const v2f*)(bptr + k);   // global_load_b64 (W, cache-resident)
        c = __builtin_amdgcn_wmma_f32_16x16x4_f32(
                /*neg_a=*/false, a, /*neg_b=*/false, b,
                /*c_mod=*/(short)0, c, /*reuse_a=*/false, /*reuse_b=*/false);
    }

    // ---- scatter logits tile to LDS: token-major rows of 16 experts ----
    // C layout: VGPR r, lanes 0-15 -> token r; lanes 16-31 -> token r+8; expert = lane&15
#pragma unroll
    for (int r = 0; r < 8; ++r) {
        const int tl = r + ((lane >> 4) << 3);
        logits_lds[wave][tl][m] = c[r];
    }
    // Same-wave LDS RAW: order + drain DS counter explicitly.
    __builtin_amdgcn_wave_barrier();
    asm volatile("s_wait_dscnt 0" ::: "memory");

    // ---- per-token top-2 / softmax / output / loss bins (lanes 0-15) ----
    if (lane < 16) {
        const float* __restrict__ lrow = &logits_lds[wave][lane][0];
        // Whole row in 4x ds_load_b128.
        v4f q0 = *(const v4f*)(lrow +  0);
        v4f q1 = *(const v4f*)(lrow +  4);
        v4f q2 = *(const v4f*)(lrow +  8);
        v4f q3 = *(const v4f*)(lrow + 12);
        float vals[16] = {q0.x, q0.y, q0.z, q0.w,
                          q1.x, q1.y, q1.z, q1.w,
                          q2.x, q2.y, q2.z, q2.w,
                          q3.x, q3.y, q3.z, q3.w};

        // Branchless top-2 scan: pure v_cmp + v_cndmask, no exec branching.
        float b0 = vals[0], b1 = -__builtin_inff();
        float e0 = 0.0f,    e1 = 0.0f;
#pragma unroll
        for (int e = 1; e < 16; ++e) {
            const float v  = vals[e];
            const float fe = (float)e;
            const bool gt0 = v > b0;
            const bool gt1 = v > b1;
            const float nb1 = gt0 ? b0 : (gt1 ? v  : b1);
            const float ne1 = gt0 ? e0 : (gt1 ? fe : e1);
            b1 = nb1;  e1 = ne1;
            b0 = gt0 ? v  : b0;
            e0 = gt0 ? fe : e0;
        }

        // 2-way softmax (others are -inf): p0 = 1/(1+e^{b1-b0})
        const float t   = __expf(b1 - b0);
        const float inv = 1.0f / (1.0f + t);
        const float p0  = inv;
        const float p1  = t * inv;

        const int tok = tile_base + lane;
        *(v2f*)(out_idx   + (size_t)tok * 2) = (v2f){e0, e1};
        *(v2f*)(out_score + (size_t)tok * 2) = (v2f){p0, p1};

        const int ie0 = (int)e0, ie1 = (int)e1;
        atomicAdd(&cnt_lds[ie0], 1.0f);     // ds_add_f32
        atomicAdd(&cnt_lds[ie1], 1.0f);
        atomicAdd(&psum_lds[ie0], p0);
        atomicAdd(&psum_lds[ie1], p1);
    }

    __syncthreads();
    if (tid < N_EXPERT_C) {
        atomicAdd(&ws_cnt[tid],  cnt_lds[tid]);    // global_atomic_add_f32
        atomicAdd(&ws_psum[tid], psum_lds[tid]);
    }
}

// ---------------------------------------------------------------------------
// Kernel 2: loss = E/N^2 * sum_e cnt[e] * psum[e]   (single wave)
// ---------------------------------------------------------------------------
__global__ void moe_loss_kernel(const float* __restrict__ ws_cnt,
                                const float* __restrict__ ws_psum,
                                float* __restrict__ loss_out)
{
    const int lane = threadIdx.x & 31;
    float v = 0.0f;
    if (lane < N_EXPERT_C) v = ws_cnt[lane] * ws_psum[lane];
#pragma unroll
    for (int off = 8; off > 0; off >>= 1) v += __shfl_down(v, off, 32);
    if (lane == 0) {
        const float scale = (float)N_EXPERT_C /
                            ((float)N_TOKENS_C * (float)N_TOKENS_C);
        *loss_out = v * scale;
    }
}

// ---------------------------------------------------------------------------
extern "C" void kernel_launch(void* const* d_in, const int* in_sizes, int n_in,
                              void* d_out, int out_size, void* d_ws, size_t ws_size,
                              hipStream_t stream) {
    const float* inp = (const float*)d_in[0];   // [32768, 1024] f32
    const float* W   = (const float*)d_in[1];   // [16, 1024] f32

    float* out       = (float*)d_out;
    float* out_idx   = out;                      // [N,2] indices (as float)
    float* out_score = out + 2 * N_TOKENS_C;     // [N,2] scores
    float* loss_out  = out + 4 * N_TOKENS_C;     // scalar loss

    float* ws      = (float*)d_ws;               // 32 floats scratch
    float* ws_cnt  = ws;
    float* ws_psum = ws + N_EXPERT_C;

    moe_init_ws<<<1, 32, 0, stream>>>(ws);

    const int blocks = N_TOKENS_C / TOKENS_PER_BLOCK;  // 256
    moe_gate_kernel<<<blocks, 256, 0, stream>>>(inp, W, out_idx, out_score,
                                                ws_cnt, ws_psum);

    moe_loss_kernel<<<1, 32, 0, stream>>>(ws_cnt, ws_psum, loss_out);
}